// RSLDS_16174846836750
// MI455X (gfx1250) — compile-verified
//
#include <hip/hip_runtime.h>

// Problem constants (from the reference)
#define BB 8
#define TT 1024
#define NN 64
#define KK 8
#define CHAINS (BB * NN)          // 512 independent (b,n) chains
#define WPB 8                     // waves per block (1 chain per wave)
#define TILE 16                   // timesteps staged per async tile
#define NTILES (TT / TILE)        // 64

// -------- wave32 logsumexp helpers (lane l owns (i=l>>3, j=l&7) and (i+4, j)) ------

// logsumexp over the 8 lanes of a j-group (xor 1,2,4); result replicated in group
__device__ __forceinline__ float lse_over_j(float x) {
    float m = x;
    m = fmaxf(m, __shfl_xor(m, 1, 32));
    m = fmaxf(m, __shfl_xor(m, 2, 32));
    m = fmaxf(m, __shfl_xor(m, 4, 32));
    float s = __expf(x - m);
    s += __shfl_xor(s, 1, 32);
    s += __shfl_xor(s, 2, 32);
    s += __shfl_xor(s, 4, 32);
    return m + __logf(s);
}

// logsumexp over the 8 i-values held as (lo,hi) across the 4 groups (xor 8,16);
// result replicated across the 4 lanes of the column (lane&7 fixed)
__device__ __forceinline__ float lse_over_i(float xlo, float xhi) {
    float m = fmaxf(xlo, xhi);
    m = fmaxf(m, __shfl_xor(m, 8, 32));
    m = fmaxf(m, __shfl_xor(m, 16, 32));
    float s = __expf(xlo - m) + __expf(xhi - m);
    s += __shfl_xor(s, 8, 32);
    s += __shfl_xor(s, 16, 32);
    return m + __logf(s);
}

// logsumexp over all 64 elements (both regs, full wave)
__device__ __forceinline__ float lse_all64(float xlo, float xhi) {
    float m = fmaxf(xlo, xhi);
    m = fmaxf(m, __shfl_xor(m, 1, 32));
    m = fmaxf(m, __shfl_xor(m, 2, 32));
    m = fmaxf(m, __shfl_xor(m, 4, 32));
    m = fmaxf(m, __shfl_xor(m, 8, 32));
    m = fmaxf(m, __shfl_xor(m, 16, 32));
    float s = __expf(xlo - m) + __expf(xhi - m);
    s += __shfl_xor(s, 1, 32);
    s += __shfl_xor(s, 2, 32);
    s += __shfl_xor(s, 4, 32);
    s += __shfl_xor(s, 8, 32);
    s += __shfl_xor(s, 16, 32);
    return m + __logf(s);
}

__device__ __forceinline__ void wait_async_le9() {
    asm volatile("s_wait_asynccnt 0x9" ::: "memory");   // older tile (9 ops) landed
}
__device__ __forceinline__ void wait_async_0() {
    asm volatile("s_wait_asynccnt 0x0" ::: "memory");   // full drain
}

__global__ __launch_bounds__(256, 1)
void rslds_fb_kernel(const float* __restrict__ log_a,
                     const float* __restrict__ log_b,
                     const float* __restrict__ z1,
                     float* __restrict__ out) {
    // 8 waves x double-buffered tiles: log_a 16*64 floats (4KB) + log_b 16*8 floats (512B)
    __shared__ float lds_a[WPB][2][TILE * KK * KK];   // 64 KB
    __shared__ float lds_b[WPB][2][TILE * KK];        //  8 KB

    const int tid  = threadIdx.x;
    const int w    = tid >> 5;
    const int lane = tid & 31;
    const int chain = blockIdx.x * WPB + w;
    const int bb = chain / NN;
    const int nn = chain % NN;

    const int jj  = lane & 7;      // column
    const int gi  = lane >> 3;     // group 0..3
    const int ilo = gi;            // row (low register)
    const int ihi = gi + 4;        // row (high register)
    const int src = (jj & 3) * 8;  // lane holding state jj's value in lo/hi regs
    const int gb8 = lane & 24;     // group base lane

    const size_t aT = (size_t)NN * KK * KK;                     // 4096 floats per t
    const size_t bT = (size_t)NN * KK;                          // 512 floats per t
    const size_t a_base  = ((size_t)bb * TT * NN + nn) * (KK * KK);
    const size_t b_base  = ((size_t)bb * TT * NN + nn) * KK;
    const size_t io_base = b_base;                              // fwd/bwd/g1 layout == log_b
    const size_t g2_base = a_base;                              // g2 layout == log_a

    float* fwd = out;
    float* bwd = out + (size_t)BB * TT * NN * KK;
    float* g1  = out + 2 * (size_t)BB * TT * NN * KK;
    float* g2  = out + 3 * (size_t)BB * TT * NN * KK;

    // LDS byte offsets (low 32 bits of a generic LDS pointer == LDS address)
    const unsigned lds_wa = (unsigned)(size_t)(&lds_a[w][0][0]);
    const unsigned lds_wb = (unsigned)(size_t)(&lds_b[w][0][0]);
    const unsigned long long a_bytes = (unsigned long long)(size_t)log_a + a_base * 4ull;
    const unsigned long long b_bytes = (unsigned long long)(size_t)log_b + b_base * 4ull;

    // Stage one 16-step tile: log_a = 16 x 256B chunks as 8 async b128 ops
    // (pair p moves steps (2p,2p+1): lanes 0-15 -> 2p, 16-31 -> 2p+1);
    // log_b = 16 x 32B chunks as 1 async b128 op (lane l -> step l>>1, half l&1).
    auto issue_tile = [&](int tile, int buf) {
        const unsigned laa = lds_wa + (unsigned)buf * (TILE * KK * KK * 4)
                                    + (unsigned)(lane * 16);
        const unsigned long long gaa =
            a_bytes + (unsigned long long)(tile * TILE + (lane >> 4)) * (aT * 4)
                    + (unsigned)((lane & 15) * 16);
#pragma unroll
        for (int p = 0; p < 8; ++p) {
            unsigned la = laa + (unsigned)(p * 512);
            unsigned long long ga = gaa + (unsigned long long)p * 2ull * aT * 4ull;
            asm volatile("global_load_async_to_lds_b128 %0, %1, off"
                         :: "v"(la), "v"(ga) : "memory");
        }
        {
            unsigned la = lds_wb + (unsigned)buf * (TILE * KK * 4)
                                 + (unsigned)(lane * 16);
            unsigned long long ga =
                b_bytes + (unsigned long long)(tile * TILE + (lane >> 1)) * (bT * 4)
                        + (unsigned)((lane & 1) * 16);
            asm volatile("global_load_async_to_lds_b128 %0, %1, off"
                         :: "v"(la), "v"(ga) : "memory");
        }
    };

    // ============================ FORWARD ============================
    // t = 0: alpha0 = normalize(z1 + b0)
    float u_lo = z1[nn * KK + ilo] + log_b[b_base + ilo];
    float u_hi = z1[nn * KK + ihi] + log_b[b_base + ihi];
    float nrm0 = lse_over_i(u_lo, u_hi);
    float al0 = u_lo - nrm0, ah0 = u_hi - nrm0;
    float plo = __shfl(al0, src, 32);
    float phi = __shfl(ah0, src, 32);
    float prev = (jj < 4) ? plo : phi;             // alpha_{t-1}[jj]
    if (lane < 8) fwd[io_base + lane] = prev;      // jj == lane for lane < 8

    issue_tile(0, 0);
    issue_tile(1, 1);
    for (int g = 0; g < NTILES; ++g) {
        const int buf = g & 1;
        if (g + 1 < NTILES) wait_async_le9(); else wait_async_0();
        const float* cha = &lds_a[w][buf][0];
        const float* chb = &lds_b[w][buf][0];
#pragma unroll 2
        for (int tt = 0; tt < TILE; ++tt) {
            const int t = g * TILE + tt;
            if (t == 0) continue;
            float a_lo = cha[tt * 64 + lane];        // a_t[ilo, jj]
            float a_hi = cha[tt * 64 + 32 + lane];   // a_t[ihi, jj]
            float bl = chb[tt * 8 + ilo];
            float bh = chb[tt * 8 + ihi];
            float ul = lse_over_j(a_lo + prev) + bl;
            float uh = lse_over_j(a_hi + prev) + bh;
            float nr = lse_over_i(ul, uh);
            float nal = ul - nr, nah = uh - nr;
            float pl = __shfl(nal, src, 32);
            float ph = __shfl(nah, src, 32);
            prev = (jj < 4) ? pl : ph;
            if (lane < 8) fwd[io_base + (size_t)t * bT + lane] = prev;
        }
        if (g + 2 < NTILES) issue_tile(g + 2, buf);
    }

    // ================= BACKWARD (fused beta / gamma1 / gamma2) =================
    if (lane < 8) bwd[io_base + (size_t)(TT - 1) * bT + lane] = 0.0f;   // beta_{T-1} = 0
    float beta_lo = 0.0f, beta_hi = 0.0f;                               // beta_t[ilo], beta_t[ihi]
    float acur = fwd[io_base + (size_t)(TT - 1) * bT + jj];             // alpha_t[jj]

    issue_tile(NTILES - 1, (NTILES - 1) & 1);
    issue_tile(NTILES - 2, (NTILES - 2) & 1);
    for (int g = NTILES - 1; g >= 0; --g) {
        const int buf = g & 1;
        if (g > 0) wait_async_le9(); else wait_async_0();
        const float* cha = &lds_a[w][buf][0];
        const float* chb = &lds_b[w][buf][0];
#pragma unroll 2
        for (int tt = TILE - 1; tt >= 0; --tt) {
            const int t = g * TILE + tt;
            if (t == 0) break;                         // loop covers t = T-1 .. 1
            float a_lo = cha[tt * 64 + lane];
            float a_hi = cha[tt * 64 + 32 + lane];
            float bl = chb[tt * 8 + ilo];
            float bh = chb[tt * 8 + ihi];

            // gamma1[t] = normalize(alpha_t + beta_t)
            float at_lo = __shfl(acur, gb8 | ilo, 32);
            float at_hi = __shfl(acur, gb8 | ihi, 32);
            float gl = at_lo + beta_lo, gh = at_hi + beta_hi;
            float gn = lse_over_i(gl, gh);
            float g1pl = __shfl(gl - gn, src, 32);
            float g1ph = __shfl(gh - gn, src, 32);
            float g1v = (jj < 4) ? g1pl : g1ph;
            if (lane < 8) g1[io_base + (size_t)t * bT + lane] = g1v;

            // alpha_{t-1} (L2-resident, off the serial chain)
            float aprev = fwd[io_base + (size_t)(t - 1) * bT + jj];

            // gamma2[t][i,j] = alpha_{t-1}[j] + beta_t[i] + b_t[i] + a_t[i,j] - lse_all
            float q_lo = aprev + beta_lo + bl + a_lo;
            float q_hi = aprev + beta_hi + bh + a_hi;
            float qn = lse_all64(q_lo, q_hi);
            const size_t g2o = g2_base + (size_t)t * aT;
            g2[g2o + lane]      = q_lo - qn;           // element (ilo, jj) == flat lane
            g2[g2o + 32 + lane] = q_hi - qn;           // element (ihi, jj) == flat lane+32

            // beta_{t-1}[j] = normalize_j( lse_i( beta_t[i] + a_t[i,j] + b_t[i] ) )
            float uj = lse_over_i(beta_lo + a_lo + bl, beta_hi + a_hi + bh);
            float bn = lse_over_j(uj);
            float bnew = uj - bn;                      // beta_{t-1}[jj]
            if (lane < 8) bwd[io_base + (size_t)(t - 1) * bT + lane] = bnew;
            beta_lo = __shfl(bnew, gb8 | ilo, 32);
            beta_hi = __shfl(bnew, gb8 | ihi, 32);
            acur = aprev;
        }
        if (g - 2 >= 0) issue_tile(g - 2, buf);
    }

    // t = 0 epilogue: gamma1[0], gamma2[0] = 0
    {
        float at_lo = __shfl(acur, gb8 | ilo, 32);     // acur == alpha_0[jj]
        float at_hi = __shfl(acur, gb8 | ihi, 32);
        float gl = at_lo + beta_lo, gh = at_hi + beta_hi;
        float gn = lse_over_i(gl, gh);
        float g1pl = __shfl(gl - gn, src, 32);
        float g1ph = __shfl(gh - gn, src, 32);
        float g1v = (jj < 4) ? g1pl : g1ph;
        if (lane < 8) g1[io_base + lane] = g1v;
        g2[g2_base + lane]      = 0.0f;
        g2[g2_base + 32 + lane] = 0.0f;
    }
}

extern "C" void kernel_launch(void* const* d_in, const int* in_sizes, int n_in,
                              void* d_out, int out_size, void* d_ws, size_t ws_size,
                              hipStream_t stream) {
    (void)in_sizes; (void)n_in; (void)d_ws; (void)ws_size; (void)out_size;
    const float* log_a = (const float*)d_in[0];   // (B,T,N,K,K)
    const float* log_b = (const float*)d_in[1];   // (B,T,N,K)
    const float* z1    = (const float*)d_in[2];   // (N,K)
    float* out = (float*)d_out;                   // fwd | bwd | gamma1 | gamma2 (flat)
    dim3 grid(CHAINS / WPB);                      // 64 blocks
    dim3 block(WPB * 32);                         // 256 threads = 8 waves
    rslds_fb_kernel<<<grid, block, 0, stream>>>(log_a, log_b, z1, out);
}